// PartialLoss_19963007992276
// MI455X (gfx1250) — compile-verified
//
#include <hip/hip_runtime.h>
#include <stdint.h>

// Problem constants from the reference: predict (B=4, C=19, H=1024, W=1024) f32,
// target (4,1024,1024) i32, loc_* are int64[N].
static constexpr int        kC  = 19;
static constexpr int        kW  = 1024;
static constexpr long long  kHW = 1024LL * 1024LL;

// wave32 butterfly reduction (gfx1250 is wave32-only; warpSize == 32)
__device__ __forceinline__ float wave_reduce_add(float v) {
#pragma unroll
  for (int off = 16; off > 0; off >>= 1)
    v += __shfl_down(v, off, 32);
  return v;
}

__global__ __launch_bounds__(256) void partial_nll_kernel(
    const float* __restrict__ predict,
    const int* __restrict__ target,
    const long long* __restrict__ loc_b,
    const long long* __restrict__ loc_row,
    const long long* __restrict__ loc_col,
    float* __restrict__ partial,
    int N)
{
  __shared__ float wsums[8];

  const int i = blockIdx.x * blockDim.x + threadIdx.x;
  float nll = 0.0f;

  if (i < N) {
    const long long b   = loc_b[i];
    const long long pix = loc_row[i] * (long long)kW + loc_col[i];

    const float* p = predict + (size_t)(b * kC) * (size_t)kHW + (size_t)pix;
    int lbl = target[(size_t)(b * kHW + pix)];
    lbl = lbl < 0 ? 0 : (lbl >= kC ? kC - 1 : lbl);

    // Issue all 19 scattered channel loads back-to-back; the compiler batches
    // them under LOADcnt and emits a single s_wait_loadcnt before first use.
    float x[kC];
#pragma unroll
    for (int ch = 0; ch < kC; ++ch)
      x[ch] = p[(size_t)ch * (size_t)kHW];

    // softmax over channels (max-subtracted)
    float m = x[0];
#pragma unroll
    for (int ch = 1; ch < kC; ++ch) m = fmaxf(m, x[ch]);

    float S = 0.0f, e_lbl = 0.0f;
#pragma unroll
    for (int ch = 0; ch < kC; ++ch) {
      const float e = __expf(x[ch] - m);   // v_exp_f32
      x[ch] = e;
      S += e;
      if (ch == lbl) e_lbl = e;            // unrolled -> v_cndmask chain, x[] stays in VGPRs
    }
    const float inv = __builtin_amdgcn_rcpf(S);  // 1/S; max(softmax) == inv since max(e)==1

    // log_softmax of the softmax vector, stabilized with its exact max (== inv)
    float S2 = 0.0f;
#pragma unroll
    for (int ch = 0; ch < kC; ++ch) S2 += __expf(x[ch] * inv - inv);

    nll = __logf(S2) + inv - e_lbl * inv;  // -(s_lbl - lse2)
  }

  // deterministic block reduction: wave32 shuffle tree -> LDS -> wave 0
  const int lane = threadIdx.x & 31;
  const int wid  = threadIdx.x >> 5;
  float v = wave_reduce_add(nll);
  if (lane == 0) wsums[wid] = v;
  __syncthreads();
  if (wid == 0) {
    float w = (lane < 8) ? wsums[lane] : 0.0f;
    w = wave_reduce_add(w);
    if (lane == 0) partial[blockIdx.x] = w;
  }
}

__global__ __launch_bounds__(256) void finalize_kernel(
    const float* __restrict__ partial, int nparts,
    float* __restrict__ out, float invN)
{
  __shared__ float wsums[8];
  float acc = 0.0f;
  for (int i = threadIdx.x; i < nparts; i += 256) acc += partial[i];

  const int lane = threadIdx.x & 31;
  const int wid  = threadIdx.x >> 5;
  float v = wave_reduce_add(acc);
  if (lane == 0) wsums[wid] = v;
  __syncthreads();
  if (wid == 0) {
    float w = (lane < 8) ? wsums[lane] : 0.0f;
    w = wave_reduce_add(w);
    if (lane == 0) out[0] = w * invN;
  }
}

extern "C" void kernel_launch(void* const* d_in, const int* in_sizes, int n_in,
                              void* d_out, int out_size, void* d_ws, size_t ws_size,
                              hipStream_t stream) {
  const float*     predict = (const float*)d_in[0];
  const int*       target  = (const int*)d_in[1];
  const long long* loc_b   = (const long long*)d_in[2];   // int64 per reference
  const long long* loc_row = (const long long*)d_in[3];
  const long long* loc_col = (const long long*)d_in[4];

  const int N       = in_sizes[2];           // 500000
  const int threads = 256;                   // 8 wave32s per block
  const int nblocks = (N + threads - 1) / threads;

  float* partial = (float*)d_ws;             // nblocks floats (~8 KB) of scratch

  partial_nll_kernel<<<nblocks, threads, 0, stream>>>(
      predict, target, loc_b, loc_row, loc_col, partial, N);
  finalize_kernel<<<1, threads, 0, stream>>>(
      partial, nblocks, (float*)d_out, 1.0f / (float)N);
}